// MKG_AttnEdgeSamp_90366111908552
// MI455X (gfx1250) — compile-verified
//
#include <hip/hip_runtime.h>

#define N_ENTITIES 200000
#define N_ITEMS    100000
#define N_REL      40
#define DIM        64
#define N_EDGES    1500000
#define K_EDGES    256
#define K_ITEMS    100
#define SCALE_LOGIT 0.08838834764831845f  /* 1/(2*sqrt(32)) : /sqrt(Dk) then mean over 2 heads */

typedef float v2f __attribute__((ext_vector_type(2)));
typedef float v8f __attribute__((ext_vector_type(8)));

// ---------- order-preserving float <-> uint mapping (for atomic max / radix top-k) ----------
__device__ __forceinline__ unsigned fmap(float f) {
  unsigned b = __float_as_uint(f);
  return (b & 0x80000000u) ? ~b : (b | 0x80000000u);
}
__device__ __forceinline__ float funmap(unsigned m) {
  unsigned b = (m & 0x80000000u) ? (m & 0x7FFFFFFFu) : ~m;
  return __uint_as_float(b);
}

// ---------- generic zeroing ----------
__global__ void zero_u32_kernel(unsigned* __restrict__ p, long n) {
  long i = (long)blockIdx.x * blockDim.x + threadIdx.x;
  long stride = (long)gridDim.x * blockDim.x;
  for (; i < n; i += stride) p[i] = 0u;
}

// ---------- 1) proj = entity_emb @ W_Q via V_WMMA_F32_16X16X4_F32 ----------
// Block = 128 threads (4 waves). Block handles 16 rows; wave w handles column tile w (16 cols).
// ISA layouts (CDNA5 ISA):
//   A 16x4 f32 : lanes 0-15 -> M=lane,   VGPR0=K0, VGPR1=K1 ; lanes 16-31 -> M=lane-16, K2/K3
//   B 4x16 f32 : lanes 0-15 -> N=lane,   VGPR0=K0, VGPR1=K1 ; lanes 16-31 -> N=lane-16, K2/K3
//   C 16x16 f32: VGPR r -> M=r (lanes 0-15) / M=8+r (lanes 16-31), N=lane%16
__global__ __launch_bounds__(128) void gemm_proj_wmma(
    const float* __restrict__ A, const float* __restrict__ W,
    float* __restrict__ P, int M) {
  const int tileM  = blockIdx.x;
  const int wave   = threadIdx.x >> 5;
  const int lane   = threadIdx.x & 31;
  const int laneLo = lane & 15;
  const int kHi    = (lane >> 4) * 2;       // 0 for lanes 0-15, 2 for lanes 16-31

  const int row = tileM * 16 + laneLo;
  const int col = wave * 16 + laneLo;
  const float* __restrict__ arow = A + (size_t)row * DIM;

  v8f acc = {};
#pragma unroll
  for (int k0 = 0; k0 < DIM; k0 += 4) {
    v2f a, b;
    a.x = arow[k0 + kHi + 0];
    a.y = arow[k0 + kHi + 1];
    b.x = W[(size_t)(k0 + kHi + 0) * DIM + col];
    b.y = W[(size_t)(k0 + kHi + 1) * DIM + col];
    acc = __builtin_amdgcn_wmma_f32_16x16x4_f32(false, a, false, b,
                                                (short)0, acc, false, false);
  }
  const int mbase = tileM * 16 + ((lane >> 4) ? 8 : 0);
#pragma unroll
  for (int r = 0; r < 8; ++r)
    P[(size_t)(mbase + r) * DIM + (wave * 16 + laneLo)] = acc[r];
}

// ---------- 2) per-edge logits + node-side atomics ----------
// 16 lanes per edge: lane s loads float4 (16B) at dim offset 4*s of q, k, rel.
__global__ __launch_bounds__(256) void edge_logits_kernel(
    const float* __restrict__ proj, const float* __restrict__ rel_emb,
    const int* __restrict__ head, const int* __restrict__ tail,
    const int* __restrict__ etype,
    float* __restrict__ logits, float* __restrict__ sum_by_node,
    unsigned* __restrict__ segmax_u, float* __restrict__ deg, int E) {
  __shared__ float srel[(N_REL - 1) * DIM];
  for (int i = threadIdx.x; i < (N_REL - 1) * DIM; i += blockDim.x)
    srel[i] = rel_emb[i];
  __syncthreads();

  const int g   = blockIdx.x * (blockDim.x >> 4) + (threadIdx.x >> 4);
  const int seg = threadIdx.x & 15;
  if (g >= E) return;

  const int h = head[g], t = tail[g], r = etype[g] - 1;
  const float4 q  = ((const float4*)(proj + (size_t)h * DIM))[seg];
  const float4 k  = ((const float4*)(proj + (size_t)t * DIM))[seg];
  const float4 rr = ((const float4*)(srel + r * DIM))[seg];
  float p = q.x * k.x * rr.x + q.y * k.y * rr.y + q.z * k.z * rr.z + q.w * k.w * rr.w;
  p += __shfl_xor(p, 8, 16);
  p += __shfl_xor(p, 4, 16);
  p += __shfl_xor(p, 2, 16);
  p += __shfl_xor(p, 1, 16);
  if (seg == 0) {
    const float logit = p * SCALE_LOGIT;
    logits[g] = logit;
    atomicAdd(&sum_by_node[h], logit);
    atomicAdd(&sum_by_node[t], logit);
    atomicAdd(&deg[h], 1.0f);
    atomicMax(&segmax_u[h], fmap(logit));
  }
}

// ---------- 3) ex = exp(logit - segmax[head]); denom accumulation ----------
__global__ void edge_exp_kernel(const float* __restrict__ logits,
                                const int* __restrict__ head,
                                const unsigned* __restrict__ segmax_u,
                                float* __restrict__ exb, float* __restrict__ denom,
                                int E) {
  int i = blockIdx.x * blockDim.x + threadIdx.x;
  if (i >= E) return;
  const int h = head[i];
  const float ex = expf(logits[i] - funmap(segmax_u[h]));
  exb[i] = ex;
  atomicAdd(&denom[h], ex);
}

// ---------- 4) score = ex/denom*deg; gumbel-noised key for top-k ----------
__global__ void edge_score_kernel(const float* __restrict__ exb,
                                  const int* __restrict__ head,
                                  const float* __restrict__ denom,
                                  const float* __restrict__ deg,
                                  const float* __restrict__ noise_u,
                                  float* __restrict__ out_score,
                                  unsigned* __restrict__ ekeys, int E) {
  int i = blockIdx.x * blockDim.x + threadIdx.x;
  if (i >= E) return;
  const int h = head[i];
  const float sc = exb[i] * deg[h] / denom[h];
  out_score[i] = sc;
  const float gum = -logf(-logf(noise_u[i]));
  ekeys[i] = fmap(sc + gum);
}

// ---------- 5) item keys from sum_by_node[:N_ITEMS] ----------
__global__ void item_keys_kernel(const float* __restrict__ sum_by_node,
                                 unsigned* __restrict__ ikeys, int n) {
  int i = blockIdx.x * blockDim.x + threadIdx.x;
  if (i < n) ikeys[i] = fmap(sum_by_node[i]);
}

// ---------- radix-select (top-K of mapped uint keys), MSB-first, 4 byte-levels ----------
// ctrl[0] = prefix, ctrl[1] = still-needed count within prefix
__global__ void sel_init_kernel(unsigned* __restrict__ ctrl, unsigned K) {
  ctrl[0] = 0u; ctrl[1] = K;
}
__global__ void radix_hist_kernel(const unsigned* __restrict__ keys, int n,
                                  const unsigned* __restrict__ ctrl, int level,
                                  unsigned* __restrict__ hist) {
  __shared__ unsigned sh[256];
  sh[threadIdx.x] = 0u;
  __syncthreads();
  const unsigned prefix = ctrl[0];
  int i = blockIdx.x * blockDim.x + threadIdx.x;
  const int stride = gridDim.x * blockDim.x;
  for (; i < n; i += stride) {
    const unsigned k = keys[i];
    const bool ok = (level == 3) || ((k >> ((level + 1) * 8)) == prefix);
    if (ok) atomicAdd(&sh[(k >> (level * 8)) & 255u], 1u);
  }
  __syncthreads();
  if (sh[threadIdx.x]) atomicAdd(&hist[threadIdx.x], sh[threadIdx.x]);
}
__global__ void radix_scan_kernel(unsigned* __restrict__ ctrl,
                                  unsigned* __restrict__ hist) {
  unsigned need = ctrl[1], cum = 0u; int done = 0;
  for (int b = 255; b >= 0; --b) {
    const unsigned h = hist[b];
    hist[b] = 0u;                       // leave hist zeroed for the next level
    if (!done) {
      if (cum + h >= need) {
        ctrl[1] = need - cum;
        ctrl[0] = (ctrl[0] << 8) | (unsigned)b;
        done = 1;
      } else cum += h;
    }
  }
}
__global__ void radix_gather_kernel(const unsigned* __restrict__ keys, int n,
                                    const unsigned* __restrict__ ctrl, int K,
                                    unsigned* __restrict__ cnt,
                                    unsigned* __restrict__ cand_key,
                                    unsigned* __restrict__ cand_id) {
  const unsigned T = ctrl[0];
  const unsigned need = ctrl[1];
  int i = blockIdx.x * blockDim.x + threadIdx.x;
  const int stride = gridDim.x * blockDim.x;
  for (; i < n; i += stride) {
    const unsigned k = keys[i];
    if (k > T) {
      const unsigned p = atomicAdd(&cnt[0], 1u);
      if (p < (unsigned)K) { cand_key[p] = k; cand_id[p] = (unsigned)i; }
    } else if (k == T) {
      const unsigned t2 = atomicAdd(&cnt[1], 1u);
      if (t2 < need) {
        const unsigned p = (unsigned)K - need + t2;
        cand_key[p] = k; cand_id[p] = (unsigned)i;
      }
    }
  }
}
// single block of 256: bitonic sort 256 candidates descending, emit first K
__global__ __launch_bounds__(256) void sort_emit_kernel(
    const unsigned* __restrict__ cand_key, const unsigned* __restrict__ cand_id,
    int K, float* __restrict__ out_v, float* __restrict__ out_id) {
  __shared__ unsigned sk[256];
  __shared__ unsigned si[256];
  const int t = threadIdx.x;
  sk[t] = cand_key[t];
  si[t] = cand_id[t];
  for (unsigned size = 2; size <= 256; size <<= 1) {
    for (unsigned stride = size >> 1; stride > 0; stride >>= 1) {
      __syncthreads();
      const unsigned j = (unsigned)t ^ stride;
      if (j > (unsigned)t) {
        const bool desc = ((t & size) == 0);
        const unsigned a = sk[t], b = sk[j];
        if (desc ? (a < b) : (a > b)) {
          sk[t] = b; sk[j] = a;
          const unsigned ia = si[t]; si[t] = si[j]; si[j] = ia;
        }
      }
    }
  }
  __syncthreads();
  if (t < K) {
    out_v[t]  = funmap(sk[t]);
    out_id[t] = (float)si[t];
  }
}

extern "C" void kernel_launch(void* const* d_in, const int* in_sizes, int n_in,
                              void* d_out, int out_size, void* d_ws, size_t ws_size,
                              hipStream_t stream) {
  const float* entity  = (const float*)d_in[0];
  const float* WQ      = (const float*)d_in[1];
  const float* rel     = (const float*)d_in[2];
  const float* noise_u = (const float*)d_in[3];
  const int*   eidx    = (const int*)d_in[4];
  const int*   etype   = (const int*)d_in[5];
  const int* head = eidx;
  const int* tail = eidx + N_EDGES;
  float* out = (float*)d_out;
  (void)in_sizes; (void)n_in; (void)out_size; (void)ws_size;

  // ---- workspace carve-up (256B aligned bumps) ----
  char* ws = (char*)d_ws;
  size_t off = 0;
  auto bump = [&](size_t bytes) -> void* {
    void* p = ws + off;
    off = (off + bytes + 255) & ~(size_t)255;
    return p;
  };
  float*    proj    = (float*)   bump((size_t)N_ENTITIES * DIM * 4);
  float*    logits  = (float*)   bump((size_t)N_EDGES * 4);
  float*    exb     = (float*)   bump((size_t)N_EDGES * 4);
  unsigned* ekeys   = (unsigned*)bump((size_t)N_EDGES * 4);
  unsigned* ikeys   = (unsigned*)bump((size_t)N_ITEMS * 4);
  // zeroed-every-call region starts here:
  size_t zbase = off;
  unsigned* segmax  = (unsigned*)bump((size_t)N_ENTITIES * 4);
  float*    denom   = (float*)   bump((size_t)N_ENTITIES * 4);
  float*    deg     = (float*)   bump((size_t)N_ENTITIES * 4);
  float*    sumn    = (float*)   bump((size_t)N_ENTITIES * 4);
  unsigned* hist    = (unsigned*)bump(256 * 4);
  unsigned* ctrlA   = (unsigned*)bump(2 * 4);
  unsigned* cntA    = (unsigned*)bump(2 * 4);
  unsigned* candAk  = (unsigned*)bump(256 * 4);
  unsigned* candAi  = (unsigned*)bump(256 * 4);
  unsigned* ctrlB   = (unsigned*)bump(2 * 4);
  unsigned* cntB    = (unsigned*)bump(2 * 4);
  unsigned* candBk  = (unsigned*)bump(256 * 4);
  unsigned* candBi  = (unsigned*)bump(256 * 4);
  const long zero_n = (long)((off - zbase) / 4);

  // ---- 0) zero the accumulator region (deterministic across graph replays) ----
  zero_u32_kernel<<<1024, 256, 0, stream>>>((unsigned*)(ws + zbase), zero_n);

  // ---- 1) WMMA GEMM: proj = entity_emb @ W_Q ----
  gemm_proj_wmma<<<N_ENTITIES / 16, 128, 0, stream>>>(entity, WQ, proj, N_ENTITIES);

  // ---- 2) edge logits + node atomics (16 lanes/edge) ----
  {
    const int groups_per_block = 256 / 16;
    const int blocks = (N_EDGES + groups_per_block - 1) / groups_per_block;
    edge_logits_kernel<<<blocks, 256, 0, stream>>>(proj, rel, head, tail, etype,
                                                   logits, sumn, segmax, deg, N_EDGES);
  }

  // ---- 3) exp / denom ----
  edge_exp_kernel<<<(N_EDGES + 255) / 256, 256, 0, stream>>>(logits, head, segmax,
                                                             exb, denom, N_EDGES);

  // ---- 4) scores (to d_out) + gumbel-noised keys ----
  edge_score_kernel<<<(N_EDGES + 255) / 256, 256, 0, stream>>>(exb, head, denom, deg,
                                                               noise_u, out, ekeys, N_EDGES);

  // ---- 5) item keys ----
  item_keys_kernel<<<(N_ITEMS + 255) / 256, 256, 0, stream>>>(sumn, ikeys, N_ITEMS);

  // ---- 6) top-256 edges via 4-level radix select + bitonic sort ----
  sel_init_kernel<<<1, 1, 0, stream>>>(ctrlA, K_EDGES);
  for (int level = 3; level >= 0; --level) {
    radix_hist_kernel<<<2048, 256, 0, stream>>>(ekeys, N_EDGES, ctrlA, level, hist);
    radix_scan_kernel<<<1, 1, 0, stream>>>(ctrlA, hist);
  }
  radix_gather_kernel<<<2048, 256, 0, stream>>>(ekeys, N_EDGES, ctrlA, K_EDGES,
                                                cntA, candAk, candAi);
  sort_emit_kernel<<<1, 256, 0, stream>>>(candAk, candAi, K_EDGES,
                                          out + N_EDGES, out + N_EDGES + K_EDGES);

  // ---- 7) top-100 items ----
  sel_init_kernel<<<1, 1, 0, stream>>>(ctrlB, K_ITEMS);
  for (int level = 3; level >= 0; --level) {
    radix_hist_kernel<<<512, 256, 0, stream>>>(ikeys, N_ITEMS, ctrlB, level, hist);
    radix_scan_kernel<<<1, 1, 0, stream>>>(ctrlB, hist);
  }
  radix_gather_kernel<<<512, 256, 0, stream>>>(ikeys, N_ITEMS, ctrlB, K_ITEMS,
                                               cntB, candBk, candBi);
  sort_emit_kernel<<<1, 256, 0, stream>>>(candBk, candBi, K_ITEMS,
                                          out + N_EDGES + 2 * K_EDGES,
                                          out + N_EDGES + 2 * K_EDGES + K_ITEMS);
}